// FraudDetectionHybrid_65481071396736
// MI455X (gfx1250) — compile-verified
//
#include <hip/hip_runtime.h>
#include <math.h>

typedef float v2f __attribute__((ext_vector_type(2)));
typedef float v4f __attribute__((ext_vector_type(4)));
typedef float v8f __attribute__((ext_vector_type(8)));

// ---------------------------------------------------------------------------
// Prep kernel: build G = [Re U; Im U] (8x4, row-major) in workspace, where
// U = M_pool * M_conv and M(z,y) = CNOT * kron(RZ(z), RY(y)) * CNOT is the
// two-qubit block the reference applies to wires (0,1). Only params[0],[1] of
// each layer touch qubit 0; the (2,3) pair and squeeze params cannot affect
// <Z0>. Direct transcription of the gate algebra - no simplification assumed.
// ---------------------------------------------------------------------------
__global__ void qcnn_prep(const float* __restrict__ conv,
                          const float* __restrict__ pool,
                          float* __restrict__ G) {
  if (threadIdx.x != 0) return;
  float Mr[2][4][4], Mi[2][4][4];
  const float* prm[2] = {conv, pool};
  for (int m = 0; m < 2; ++m) {
    float z = prm[m][0], y = prm[m][1];
    float cz = cosf(0.5f * z), sz = sinf(0.5f * z);
    float cy = cosf(0.5f * y), sy = sinf(0.5f * y);
    float RY[2][2] = {{cy, -sy}, {sy, cy}};
    // A = kron(RZ(z), RY(y)); RZ = diag(cz - i sz, cz + i sz)
    float Ar[4][4] = {}, Ai[4][4] = {};
    for (int i = 0; i < 2; ++i)
      for (int j = 0; j < 2; ++j) {
        Ar[i][j]         = cz * RY[i][j];  Ai[i][j]         = -sz * RY[i][j];
        Ar[2 + i][2 + j] = cz * RY[i][j];  Ai[2 + i][2 + j] =  sz * RY[i][j];
      }
    // M = CNOT * A * CNOT  ==  permute rows & cols with p = {0,1,3,2}
    const int p[4] = {0, 1, 3, 2};
    for (int i = 0; i < 4; ++i)
      for (int j = 0; j < 4; ++j) {
        Mr[m][i][j] = Ar[p[i]][p[j]];
        Mi[m][i][j] = Ai[p[i]][p[j]];
      }
  }
  // U = M_pool * M_conv (complex 4x4)
  for (int i = 0; i < 4; ++i)
    for (int j = 0; j < 4; ++j) {
      float sr = 0.f, si = 0.f;
      for (int k = 0; k < 4; ++k) {
        sr += Mr[1][i][k] * Mr[0][k][j] - Mi[1][i][k] * Mi[0][k][j];
        si += Mr[1][i][k] * Mi[0][k][j] + Mi[1][i][k] * Mr[0][k][j];
      }
      G[i * 4 + j]       = sr;   // rows 0..3: Re U
      G[(4 + i) * 4 + j] = si;   // rows 4..7: Im U
    }
}

// ---------------------------------------------------------------------------
// Main kernel: one wave handles 16 samples.
//   P (16x4) x G^T (4x16 padded)  via v_wmma_f32_16x16x4_f32
//   ez_b = sum_j sgn_j * Y[b,j]^2   (cols 8..15 of Y are exactly zero)
//   out  = sigmoid(ez)
// ---------------------------------------------------------------------------
__global__ __launch_bounds__(256) void qcnn_main(const float* __restrict__ x,
                                                 const float* __restrict__ G,
                                                 float* __restrict__ out,
                                                 int n) {
  const int lane = threadIdx.x & 31;
  const int wave = threadIdx.x >> 5;
  const int h    = lane >> 4;     // half-wave
  const int nn   = lane & 15;     // M (rows of A) / N (cols of B,D)
  const int base = (blockIdx.x * 8 + wave) * 16;
  if (base >= n) return;          // wave-uniform: EXEC all-ones at the WMMA

  // --- build A (one sample per lane pair {l, l+16}) -------------------------
  int s = base + nn;
  if (s >= n) s = n - 1;          // clamp, no EXEC change
  __builtin_prefetch(x + 2 * (size_t)s + 4096, 0, 1);   // global_prefetch_b8
  v2f xx = __builtin_nontemporal_load((const v2f*)x + s); // streamed once
  float x0 = xx.x, x1 = xx.y;
  float c0 = sqrtf(fmaxf(0.f, 1.f - x0 * x0));   // cos(asin(x0))
  float c1 = sqrtf(fmaxf(0.f, 1.f - x1 * x1));
  float p0 = c0 * c1, p1 = c0 * x1, p2 = x0 * c1, p3 = x0 * x1;
  v2f a;                          // A layout: VGPR v, half h -> K = v + 2h
  a.x = h ? p2 : p0;
  a.y = h ? p3 : p1;

  // --- build B = G^T padded to 4x16 ----------------------------------------
  // B layout: lane -> N = lane%16; VGPR v, half h -> K = v + 2h.
  // B[k][n] = G[n][k] for n < 8, else 0.
  int gr = (nn < 8) ? nn : 0;
  float g0 = G[gr * 4 + 2 * h];
  float g1 = G[gr * 4 + 2 * h + 1];
  v2f b;
  b.x = (nn < 8) ? g0 : 0.f;
  b.y = (nn < 8) ? g1 : 0.f;

  // --- WMMA: Y = P * G^T ----------------------------------------------------
  v8f c = {};
  v8f d = __builtin_amdgcn_wmma_f32_16x16x4_f32(
      /*neg_a=*/false, a, /*neg_b=*/false, b,
      /*c_mod=*/(short)0, c, /*reuse_a=*/false, /*reuse_b=*/false);

  // --- signed squares + 16-lane reduction per half --------------------------
  // sgn_j = +1,+1,-1,-1 repeating (Z on qubit 0 over Re/Im halves of U p).
  float sgn = ((nn >> 1) & 1) ? -1.f : 1.f;
  float e[8];
#pragma unroll
  for (int r = 0; r < 8; ++r) e[r] = sgn * d[r] * d[r];
#pragma unroll
  for (int m = 1; m <= 8; m <<= 1) {
#pragma unroll
    for (int r = 0; r < 8; ++r) e[r] += __shfl_xor(e[r], m, 32);
  }
  // Now every low-half lane holds ez[base+r] in e[r]; high half ez[base+8+r].

  // --- sigmoid + store (lanes 0 and 16 emit 8 results each) ----------------
  if (nn == 0) {
    int rowbase = base + 8 * h;
    if (base + 16 <= n) {
      v4f o0, o1;
#pragma unroll
      for (int j = 0; j < 4; ++j) o0[j] = 1.f / (1.f + __expf(-e[j]));
#pragma unroll
      for (int j = 0; j < 4; ++j) o1[j] = 1.f / (1.f + __expf(-e[4 + j]));
      __builtin_nontemporal_store(o0, (v4f*)(out + rowbase));
      __builtin_nontemporal_store(o1, (v4f*)(out + rowbase) + 1);
    } else {
#pragma unroll
      for (int j = 0; j < 8; ++j) {
        int oi = rowbase + j;
        if (oi < n) out[oi] = 1.f / (1.f + __expf(-e[j]));
      }
    }
  }
}

// ---------------------------------------------------------------------------
extern "C" void kernel_launch(void* const* d_in, const int* in_sizes, int n_in,
                              void* d_out, int out_size, void* d_ws, size_t ws_size,
                              hipStream_t stream) {
  const float* x    = (const float*)d_in[0];  // [B,2] float32
  const float* conv = (const float*)d_in[1];  // [6]
  const float* pool = (const float*)d_in[2];  // [6]
  // d_in[3] = squeeze_params: identity in the reference, unused.
  float* out = (float*)d_out;                 // [B,1] float32
  float* G   = (float*)d_ws;                  // 32 floats

  int n = in_sizes[0] / 2;                    // batch size B

  qcnn_prep<<<1, 32, 0, stream>>>(conv, pool, G);
  int blocks = (n + 127) / 128;               // 8 waves/block * 16 samples/wave
  qcnn_main<<<blocks, 256, 0, stream>>>(x, G, out, n);
}